// WWL_33225867001966
// MI455X (gfx1250) — compile-verified
//
#include <hip/hip_runtime.h>

// Continuous-WL conv, 4 layers, on MI455X (gfx1250, wave32).
// CSR (dst-sorted neighbor lists) built once, then 4 atomic-free gather
// layers. Wave layout in the gather: lane[2:0]=feature quad (float4),
// lane[4:3]=neighbor slot-group, so one global_load_b128 fetches 4 neighbor
// rows (4 cachelines) per instruction. Working set (<90MB) is L2-resident.

__global__ void wl_zero_deg(int* __restrict__ deg, int n) {
    int i = blockIdx.x * blockDim.x + threadIdx.x;
    if (i < n) deg[i] = 0;
}

__global__ void wl_count_deg(const int* __restrict__ dst, int* __restrict__ deg, int E) {
    int i = blockIdx.x * blockDim.x + threadIdx.x;
    int stride = gridDim.x * blockDim.x;
    for (int e = i; e < E; e += stride) {
        atomicAdd(&deg[dst[e]], 1);
    }
}

// Single-block exclusive scan over N degrees -> CSR offsets (+ cursor copy).
__global__ void wl_scan(const int* __restrict__ deg, int* __restrict__ off,
                        int* __restrict__ cursor, int N) {
    __shared__ int partial[1024];
    const int tid = threadIdx.x;
    const int chunk = (N + 1023) >> 10;
    const int b = tid * chunk;
    const int e = min(b + chunk, N);
    int s = 0;
    for (int i = b; i < e; ++i) s += deg[i];
    partial[tid] = s;
    __syncthreads();
    for (int d = 1; d < 1024; d <<= 1) {
        int v = 0;
        if (tid >= d) v = partial[tid - d];
        __syncthreads();
        if (tid >= d) partial[tid] += v;
        __syncthreads();
    }
    int run = (tid > 0) ? partial[tid - 1] : 0;  // exclusive base of this chunk
    for (int i = b; i < e; ++i) {
        off[i] = run;
        cursor[i] = run;
        run += deg[i];
    }
    if (tid == 1023) off[N] = partial[1023];     // total edge count
}

__global__ void wl_fill(const int* __restrict__ src, const int* __restrict__ dst,
                        int* __restrict__ cursor, int* __restrict__ nbr, int E) {
    int i = blockIdx.x * blockDim.x + threadIdx.x;
    int stride = gridDim.x * blockDim.x;
    for (int e = i; e < E; e += stride) {
        int d = dst[e];
        int pos = atomicAdd(&cursor[d], 1);
        nbr[pos] = src[e];
    }
}

// One wave per node. Each inner step processes 4 neighbors simultaneously:
// slot-group g = lane>>3 picks the neighbor (broadcast via ds_bpermute),
// feature-quad fq = lane&7 picks which float4 of the 32-float row to load
// (global_load_b128). Slot-groups are folded with a 2-step shfl_xor butterfly.
__global__ void wl_layer(const float* __restrict__ xin, int stride_in,
                         const int* __restrict__ off, const int* __restrict__ nbr,
                         float* __restrict__ outp, int N) {
    const int lane = threadIdx.x & 31;
    const int wave = blockIdx.x * (blockDim.x >> 5) + (threadIdx.x >> 5);
    if (wave >= N) return;
    const int node = wave;
    const int b = off[node];
    const int e = off[node + 1];
    const int g  = lane >> 3;    // neighbor slot-group 0..3
    const int fq = lane & 7;     // feature quad 0..7 (features 4*fq .. 4*fq+3)

    float4 acc = make_float4(0.f, 0.f, 0.f, 0.f);
    for (int k = b; k < e; k += 32) {
        int idx = 0;
        if (k + lane < e) idx = nbr[k + lane];               // coalesced id load
        if (k + 32 < e) __builtin_prefetch(&nbr[k + 32], 0, 0);  // global_prefetch_b8
        const int steps = min(32, e - k);                    // valid neighbors here
        const int t4 = (steps + 3) >> 2;
        for (int t = 0; t < t4; ++t) {
            const int slot = 4 * t + g;
            const int s = __shfl(idx, slot, 32);             // ==0 if slot>=steps (safe)
            const float4 v =
                *(const float4*)(xin + (long long)s * stride_in + 4 * fq);  // b128
            if (slot < steps) {
                acc.x += v.x; acc.y += v.y; acc.z += v.z; acc.w += v.w;
            }
        }
    }

    // Fold the 4 slot-groups: lanes {l, l^8, l^16, l^24} share the same fq.
    acc.x += __shfl_xor(acc.x, 8, 32);
    acc.y += __shfl_xor(acc.y, 8, 32);
    acc.z += __shfl_xor(acc.z, 8, 32);
    acc.w += __shfl_xor(acc.w, 8, 32);
    acc.x += __shfl_xor(acc.x, 16, 32);
    acc.y += __shfl_xor(acc.y, 16, 32);
    acc.z += __shfl_xor(acc.z, 16, 32);
    acc.w += __shfl_xor(acc.w, 16, 32);

    if (lane < 8) {  // lanes 0..7 hold quads 0..7 -> one b128 store each
        const float4 xi =
            *(const float4*)(xin + (long long)node * stride_in + 4 * lane);
        const float dinv = (e > b) ? 1.0f / (float)(e - b) : 0.0f;
        float4 r;
        r.x = 0.5f * (xi.x + acc.x * dinv);
        r.y = 0.5f * (xi.y + acc.y * dinv);
        r.z = 0.5f * (xi.z + acc.z * dinv);
        r.w = 0.5f * (xi.w + acc.w * dinv);
        *(float4*)(outp + (long long)node * 128 + 4 * lane) = r;
    }
}

extern "C" void kernel_launch(void* const* d_in, const int* in_sizes, int n_in,
                              void* d_out, int out_size, void* d_ws, size_t ws_size,
                              hipStream_t stream) {
    const float* x    = (const float*)d_in[0];
    const int*   edge = (const int*)d_in[1];   // [2, E] row-major int32
    const int N = in_sizes[0] / 32;            // 100000
    const int E = in_sizes[1] / 2;             // 1600000
    const int* src = edge;                      // edge_index[0]
    const int* dst = edge + E;                  // edge_index[1]
    float* out = (float*)d_out;                 // [N, 128]

    // Workspace (ints): deg[N] | off[N+1] | cursor[N] | nbr[E]  (~7.6MB)
    int* deg    = (int*)d_ws;
    int* off    = deg + N;
    int* cursor = off + (N + 1);
    int* nbr    = cursor + N;

    // ---- CSR build (once, amortized over 4 layers) ----
    wl_zero_deg<<<(N + 255) / 256, 256, 0, stream>>>(deg, N);
    wl_count_deg<<<2048, 256, 0, stream>>>(dst, deg, E);
    wl_scan<<<1, 1024, 0, stream>>>(deg, off, cursor, N);
    wl_fill<<<2048, 256, 0, stream>>>(src, dst, cursor, nbr, E);

    // ---- 4 gather layers: wave-per-node, 8 waves per 256-thread block ----
    const int WPB = 256 / 32;
    const int blocks = (N + WPB - 1) / WPB;
    wl_layer<<<blocks, 256, 0, stream>>>(x, 32, off, nbr, out + 0, N);
    for (int l = 1; l < 4; ++l) {
        wl_layer<<<blocks, 256, 0, stream>>>(out + (l - 1) * 32, 128,
                                             off, nbr, out + l * 32, N);
    }
}